// M11_5514738008550
// MI455X (gfx1250) — compile-verified
//
#include <hip/hip_runtime.h>
#include <math.h>

#define N_NODES 50000
#define N_EDGES 400000
#define D_NODE  128
#define D_EDGE  16
#define D_CONV  128
#define N_LAYERS 3
#define BN_EPS  1e-5f
#define LEAKY   0.01f
#define MAX_CIN 384
#define EPB     16   // edges per block in scatter kernel

typedef __attribute__((ext_vector_type(2))) float v2f;
typedef __attribute__((ext_vector_type(8))) float v8f;

// ---------------- elementwise / reduction kernels ----------------

__global__ void k_copy_x(const float* __restrict__ x, float* __restrict__ H) {
    int n = blockIdx.x;
    int c = threadIdx.x;                 // blockDim = 128
    H[(long)n * 512 + c] = x[(long)n * D_NODE + c];
}

__global__ void k_fill0(float* __restrict__ p, long n) {
    long i = (long)blockIdx.x * blockDim.x + threadIdx.x;
    long stride = (long)gridDim.x * blockDim.x;
    for (; i < n; i += stride) p[i] = 0.f;
}

// column-wise mean + rsqrt(var+eps) over `nrows` rows; C must be multiple of 32
__global__ void k_bnstats(const float* __restrict__ src, int ld, int nrows,
                          float* __restrict__ mean, float* __restrict__ rstd) {
    __shared__ float s_s[8][32];
    __shared__ float s_q[8][32];
    int tx = threadIdx.x, ty = threadIdx.y;
    int col = blockIdx.x * 32 + tx;
    float s = 0.f, q = 0.f;
    for (int r = ty; r < nrows; r += 8) {
        float v = src[(long)r * ld + col];
        s += v; q += v * v;
    }
    s_s[ty][tx] = s; s_q[ty][tx] = q;
    __syncthreads();
    if (ty == 0) {
        float S = 0.f, Q = 0.f;
        #pragma unroll
        for (int j = 0; j < 8; ++j) { S += s_s[j][tx]; Q += s_q[j][tx]; }
        float m = S / (float)nrows;
        float var = fmaxf(Q / (float)nrows - m * m, 0.f);
        mean[col] = m;
        rstd[col] = rsqrtf(var + BN_EPS);
    }
}

// x_norm = g*(H[:,0:cin]-mean)*rstd + b  (+ optional leaky)
__global__ void k_bnapply(const float* __restrict__ H, int ld, int cin,
                          const float* __restrict__ g, const float* __restrict__ b,
                          const float* __restrict__ mean, const float* __restrict__ rstd,
                          int do_leaky, float* __restrict__ xn) {
    int c = blockIdx.y * blockDim.x + threadIdx.x;
    int n = blockIdx.x;
    if (c >= cin) return;
    float v = H[(long)n * ld + c];
    v = g[c] * (v - mean[c]) * rstd[c] + b[c];
    if (do_leaky) v = (v >= 0.f) ? v : LEAKY * v;
    xn[(long)n * cin + c] = v;
}

// ---------------- fused edge embed + gather + relu + scatter-add ----------------

__global__ void k_edge_scatter(const float* __restrict__ edge_attr,
                               const long long* __restrict__ srcI,
                               const long long* __restrict__ dstI,
                               const float* __restrict__ We, const float* __restrict__ be,
                               const float* __restrict__ xn, float* __restrict__ agg,
                               int cin) {
    __shared__ float sWe[MAX_CIN * 16];
    __shared__ float sEA[EPB][16];
    __shared__ int   sSrc[EPB], sDst[EPB];
    int tid = threadIdx.x;               // 256 threads
    for (int i = tid; i < cin * 16; i += 256) sWe[i] = We[i];
    long e0 = (long)blockIdx.x * EPB;
    {
        int le = tid >> 4, k = tid & 15;
        long e = e0 + le;
        sEA[le][k] = (e < N_EDGES) ? edge_attr[e * 16 + k] : 0.f;
    }
    if (tid < EPB) {
        long e = e0 + tid;
        if (e < N_EDGES) { sSrc[tid] = (int)srcI[e]; sDst[tid] = (int)dstI[e]; }
        else             { sSrc[tid] = -1;           sDst[tid] = -1; }
    }
    __syncthreads();
    int le = tid >> 4;   // local edge 0..15
    int cl = tid & 15;   // channel lane 0..15
    int s = sSrc[le], d = sDst[le];
    if (s < 0) return;
    for (int cb = 0; cb < cin; cb += 16) {
        int c = cb + cl;
        float acc = be[c];
        #pragma unroll
        for (int k = 0; k < 16; ++k) acc += sEA[le][k] * sWe[c * 16 + k];
        float m = acc + xn[(long)s * cin + c];   // x[src] + e
        m = (m > 0.f) ? m : 0.f;                 // relu
        atomicAdd(&agg[(long)d * cin + c], m);   // segment_sum
    }
}

// ---------------- WMMA f32 GEMMs ----------------
// h1 = ((1+eps)*xn + agg) @ W1^T + b1     [N x 128], K = cin
__global__ void k_gemm1_wmma(const float* __restrict__ xn, const float* __restrict__ agg,
                             const float* __restrict__ W1, const float* __restrict__ b1,
                             const float* __restrict__ epsp, int cin,
                             float* __restrict__ h1) {
    int lane = threadIdx.x & 31;
    int wave = threadIdx.x >> 5;         // 8 waves -> col tiles 0..7
    int half = lane >> 4, lrow = lane & 15;
    int rowBase = blockIdx.x * 16;
    int colBase = wave * 16;
    float eps1 = 1.f + epsp[0];
    int row = rowBase + lrow;
    int col = colBase + lrow;
    const float* xrow = xn  + (long)row * cin;
    const float* arow = agg + (long)row * cin;
    const float* wrow = W1  + (long)col * cin;
    v8f acc = {};
    for (int k = 0; k < cin; k += 4) {
        int ka = k + 2 * half;
        v2f a, b;
        a.x = eps1 * xrow[ka]     + arow[ka];
        a.y = eps1 * xrow[ka + 1] + arow[ka + 1];
        b.x = wrow[ka];
        b.y = wrow[ka + 1];
        acc = __builtin_amdgcn_wmma_f32_16x16x4_f32(false, a, false, b,
                                                    (short)0, acc, false, false);
    }
    float bias = b1[col];
    #pragma unroll
    for (int j = 0; j < 8; ++j) {
        int r = rowBase + j + 8 * half;
        h1[(long)r * D_CONV + col] = acc[j] + bias;
    }
}

// H[:, outColBase:outColBase+128] = leaky(bn(h1)) @ W2^T + b2c, K = 128
__global__ void k_gemm2_wmma(const float* __restrict__ h1,
                             const float* __restrict__ mean, const float* __restrict__ rstd,
                             const float* __restrict__ g2, const float* __restrict__ b2,
                             const float* __restrict__ W2, const float* __restrict__ b2c,
                             float* __restrict__ H, int outColBase) {
    int lane = threadIdx.x & 31;
    int wave = threadIdx.x >> 5;
    int half = lane >> 4, lrow = lane & 15;
    int rowBase = blockIdx.x * 16;
    int colBase = wave * 16;
    int row = rowBase + lrow;
    int col = colBase + lrow;
    const float* xrow = h1 + (long)row * D_CONV;
    const float* wrow = W2 + (long)col * D_CONV;
    v8f acc = {};
    for (int k = 0; k < D_CONV; k += 4) {
        int ka = k + 2 * half;
        float v0 = g2[ka]     * (xrow[ka]     - mean[ka])     * rstd[ka]     + b2[ka];
        float v1 = g2[ka + 1] * (xrow[ka + 1] - mean[ka + 1]) * rstd[ka + 1] + b2[ka + 1];
        v2f a, b;
        a.x = (v0 >= 0.f) ? v0 : LEAKY * v0;
        a.y = (v1 >= 0.f) ? v1 : LEAKY * v1;
        b.x = wrow[ka];
        b.y = wrow[ka + 1];
        acc = __builtin_amdgcn_wmma_f32_16x16x4_f32(false, a, false, b,
                                                    (short)0, acc, false, false);
    }
    float bias = b2c[col];
    #pragma unroll
    for (int j = 0; j < 8; ++j) {
        int r = rowBase + j + 8 * half;
        H[(long)r * 512 + outColBase + col] = acc[j] + bias;
    }
}

// out[n] = dot(H[n,:512], Wf) + bf
__global__ void k_final(const float* __restrict__ H, const float* __restrict__ Wf,
                        const float* __restrict__ bf, float* __restrict__ out) {
    int tx = threadIdx.x, ty = threadIdx.y;   // (32, 8)
    int row = blockIdx.x * 8 + ty;
    if (row >= N_NODES) return;
    float s = 0.f;
    for (int c = tx; c < 512; c += 32) s += H[(long)row * 512 + c] * Wf[c];
    #pragma unroll
    for (int off = 16; off > 0; off >>= 1) s += __shfl_down(s, off, 32);
    if (tx == 0) out[row] = s + bf[0];
}

// ---------------- launcher ----------------

extern "C" void kernel_launch(void* const* d_in, const int* in_sizes, int n_in,
                              void* d_out, int out_size, void* d_ws, size_t ws_size,
                              hipStream_t stream) {
    const float* x            = (const float*)d_in[0];
    const float* edge_attr    = (const float*)d_in[1];
    const long long* edge_idx = (const long long*)d_in[2];
    const long long* srcI = edge_idx;
    const long long* dstI = edge_idx + N_EDGES;

    // workspace carve-out (floats)
    float* H     = (float*)d_ws;                        // N x 512
    float* XN    = H    + (size_t)N_NODES * 512;        // N x MAX_CIN
    float* AGG   = XN   + (size_t)N_NODES * MAX_CIN;    // N x MAX_CIN
    float* H1    = AGG  + (size_t)N_NODES * MAX_CIN;    // N x 128
    float* MEAN  = H1   + (size_t)N_NODES * D_CONV;     // 512
    float* RSTD  = MEAN + 512;                          // 512
    float* MEAN2 = RSTD + 512;                          // 128
    float* RSTD2 = MEAN2 + 128;                         // 128

    k_copy_x<<<N_NODES, 128, 0, stream>>>(x, H);

    int cin = D_NODE;
    for (int i = 0; i < N_LAYERS; ++i) {
        int base = 3 + i * 11;
        const float* bn_g = (const float*)d_in[base + 0];
        const float* bn_b = (const float*)d_in[base + 1];
        const float* epsp = (const float*)d_in[base + 2];
        const float* We   = (const float*)d_in[base + 3];
        const float* be   = (const float*)d_in[base + 4];
        const float* W1   = (const float*)d_in[base + 5];
        const float* b1   = (const float*)d_in[base + 6];
        const float* g2   = (const float*)d_in[base + 7];
        const float* b2   = (const float*)d_in[base + 8];
        const float* W2   = (const float*)d_in[base + 9];
        const float* b2c  = (const float*)d_in[base + 10];

        // BN over current concat features
        k_bnstats<<<cin / 32, dim3(32, 8), 0, stream>>>(H, 512, N_NODES, MEAN, RSTD);
        dim3 gA(N_NODES, (cin + 255) / 256);
        k_bnapply<<<gA, 256, 0, stream>>>(H, 512, cin, bn_g, bn_b, MEAN, RSTD,
                                          (i > 0) ? 1 : 0, XN);
        // aggregate messages
        k_fill0<<<2048, 256, 0, stream>>>(AGG, (long)N_NODES * cin);
        k_edge_scatter<<<(N_EDGES + EPB - 1) / EPB, 256, 0, stream>>>(
            edge_attr, srcI, dstI, We, be, XN, AGG, cin);
        // nn: Linear -> BN -> LeakyReLU -> Linear (WMMA f32)
        k_gemm1_wmma<<<N_NODES / 16, 256, 0, stream>>>(XN, AGG, W1, b1, epsp, cin, H1);
        k_bnstats<<<D_CONV / 32, dim3(32, 8), 0, stream>>>(H1, D_CONV, N_NODES, MEAN2, RSTD2);
        k_gemm2_wmma<<<N_NODES / 16, 256, 0, stream>>>(H1, MEAN2, RSTD2, g2, b2, W2, b2c,
                                                       H, D_NODE + i * D_CONV);
        cin += D_CONV;
    }

    const float* Wf = (const float*)d_in[3 + N_LAYERS * 11 + 0];
    const float* bf = (const float*)d_in[3 + N_LAYERS * 11 + 1];
    k_final<<<(N_NODES + 7) / 8, dim3(32, 8), 0, stream>>>(H, Wf, bf, (float*)d_out);
}